// GATDecoder_85718957293592
// MI455X (gfx1250) — compile-verified
//
#include <hip/hip_runtime.h>

// ---------------------------------------------------------------------------
// GATConv x2 for MI455X (gfx1250, wave32).
//   GEMMs via V_WMMA_F32_16X16X4_F32 (f32 A/B, full precision).
//   Edge softmax/aggregation via global f32 atomics (hits 192MB L2).
// ---------------------------------------------------------------------------

typedef float v2f __attribute__((ext_vector_type(2)));
typedef float v8f __attribute__((ext_vector_type(8)));

#define NEG_SLOPE 0.2f

__device__ __forceinline__ void atomicMaxF(float* addr, float val) {
  // Monotone bit-pattern trick: positives ordered as int, negatives reversed as uint.
  if (val >= 0.0f) atomicMax((int*)addr, __float_as_int(val));
  else             atomicMin((unsigned int*)addr, __float_as_uint(val));
}

// ---- GEMM: H[n,FOUT] = X[n,64] @ W[64,FOUT], one wave per 16-row tile ------
template <int FOUT>
__global__ void gat_gemm_wmma(const float* __restrict__ X,
                              const float* __restrict__ W,
                              float* __restrict__ Hout, int n) {
  const int lane = threadIdx.x & 31;
  const int wave = (int)((blockIdx.x * blockDim.x + threadIdx.x) >> 5);
  const int row0 = wave * 16;
  if (row0 >= n) return;                 // wave-uniform exit (EXEC all-1 for WMMA)

  const int r  = lane & 15;              // row (A) / col (B,C,D) within tile
  const int kh = lane >> 4;              // K-half: lanes 0-15 hold K,K+1; 16-31 hold K+2,K+3

  int arow = row0 + r; if (arow >= n) arow = n - 1;
  const float* __restrict__ xrow = X + (size_t)arow * 64;

  constexpr int NT = FOUT / 16;
  v8f zero = {};
  v8f acc[NT];
#pragma unroll
  for (int t = 0; t < NT; ++t) acc[t] = zero;

#pragma unroll
  for (int k = 0; k < 64; k += 4) {
    v2f a;                                // A: 16x4 f32 tile
    a.x = xrow[k + 2 * kh];
    a.y = xrow[k + 2 * kh + 1];
#pragma unroll
    for (int t = 0; t < NT; ++t) {
      v2f b;                              // B: 4x16 f32 tile (rows striped across lanes)
      b.x = W[(size_t)(k + 2 * kh)     * FOUT + t * 16 + r];
      b.y = W[(size_t)(k + 2 * kh + 1) * FOUT + t * 16 + r];
      acc[t] = __builtin_amdgcn_wmma_f32_16x16x4_f32(
          false, a, false, b, (short)0, acc[t], false, false);
    }
  }

  // C/D layout: VGPR i -> M=i (lanes 0-15), M=i+8 (lanes 16-31); N = t*16 + r
#pragma unroll
  for (int t = 0; t < NT; ++t) {
#pragma unroll
    for (int i = 0; i < 8; ++i) {
      int row = row0 + i + 8 * kh;
      if (row < n) Hout[(size_t)row * FOUT + t * 16 + r] = acc[t][i];
    }
  }
}

// ---- per-node attention scalars: a_src[i,h] = <H[i,h,:], att_src[h,:]> -----
__global__ void attn_scalars(const float* __restrict__ H,
                             const float* __restrict__ att_src,
                             const float* __restrict__ att_dst,
                             float* __restrict__ a_src,
                             float* __restrict__ a_dst,
                             int n, int C) {
  int i = blockIdx.x * blockDim.x + threadIdx.x;
  if (i >= n) return;
  const float* hrow = H + (size_t)i * 2 * C;
  for (int h = 0; h < 2; ++h) {
    float s = 0.f, d = 0.f;
    for (int c = 0; c < C; ++c) {
      float v = hrow[h * C + c];
      s += v * att_src[h * C + c];
      d += v * att_dst[h * C + c];
    }
    a_src[i * 2 + h] = s;
    a_dst[i * 2 + h] = d;
  }
}

__device__ __forceinline__ void edge_nodes(const int* __restrict__ ei, int e,
                                           int E, int& s, int& d) {
  if (e < E) { s = ei[e]; d = ei[E + e]; }     // given edges
  else       { s = e - E; d = e - E; }         // appended self-loops
}

__device__ __forceinline__ float leaky(float a) {
  return (a > 0.f) ? a : NEG_SLOPE * a;
}

__global__ void edge_max_k(const int* __restrict__ ei,
                           const float* __restrict__ a_src,
                           const float* __restrict__ a_dst,
                           float* __restrict__ amax, int E, int EE) {
  int e = blockIdx.x * blockDim.x + threadIdx.x;
  if (e >= EE) return;
  int s, d; edge_nodes(ei, e, E, s, d);
#pragma unroll
  for (int h = 0; h < 2; ++h) {
    float al = leaky(a_src[s * 2 + h] + a_dst[d * 2 + h]);
    atomicMaxF(&amax[d * 2 + h], al);
  }
}

__global__ void edge_exp_k(const int* __restrict__ ei,
                           const float* __restrict__ a_src,
                           const float* __restrict__ a_dst,
                           const float* __restrict__ amax,
                           float* __restrict__ exA,
                           float* __restrict__ denom, int E, int EE) {
  int e = blockIdx.x * blockDim.x + threadIdx.x;
  if (e >= EE) return;
  int s, d; edge_nodes(ei, e, E, s, d);
#pragma unroll
  for (int h = 0; h < 2; ++h) {
    float al = leaky(a_src[s * 2 + h] + a_dst[d * 2 + h]);
    float ex = __expf(al - amax[d * 2 + h]);
    exA[(size_t)e * 2 + h] = ex;
    atomicAdd(&denom[d * 2 + h], ex);
  }
}

// ---- out[dst,h,:] += (exA/denom) * H[src,h,:] ; float4 per thread ----------
template <int FOUT, int C>
__global__ void edge_aggr_k(const int* __restrict__ ei,
                            const float* __restrict__ Hfeat,
                            const float* __restrict__ exA,
                            const float* __restrict__ denom,
                            float* __restrict__ out, int E, int EE) {
  constexpr int QPE = FOUT / 4;
  long long tid = (long long)blockIdx.x * blockDim.x + threadIdx.x;
  int e = (int)(tid / QPE);
  if (e >= EE) return;
  int q   = (int)(tid % QPE);
  int col = q * 4;
  int h   = col / C;
  int s, d; edge_nodes(ei, e, E, s, d);
  float w = exA[(size_t)e * 2 + h] / (denom[d * 2 + h] + 1e-16f);
  const float4 v = *(const float4*)(Hfeat + (size_t)s * FOUT + col);
  float* o = out + (size_t)d * FOUT + col;
  atomicAdd(o + 0, w * v.x);
  atomicAdd(o + 1, w * v.y);
  atomicAdd(o + 2, w * v.z);
  atomicAdd(o + 3, w * v.w);
}

__global__ void fill_k(float* __restrict__ p, float v, int n) {
  int i = blockIdx.x * blockDim.x + threadIdx.x;
  if (i < n) p[i] = v;
}

__global__ void init_bias_k(float* __restrict__ out,
                            const float* __restrict__ bias, int n, int F) {
  int i = blockIdx.x * blockDim.x + threadIdx.x;
  if (i < n * F) out[i] = bias[i % F];
}

__global__ void finalize_mean_k(const float* __restrict__ out2,
                                const float* __restrict__ b2,
                                float* __restrict__ y, int n) {
  int i = blockIdx.x * blockDim.x + threadIdx.x;
  if (i >= n * 64) return;
  int node = i >> 6, c = i & 63;
  y[i] = 0.5f * (out2[(size_t)node * 128 + c] + out2[(size_t)node * 128 + 64 + c]) + b2[c];
}

extern "C" void kernel_launch(void* const* d_in, const int* in_sizes, int n_in,
                              void* d_out, int out_size, void* d_ws, size_t ws_size,
                              hipStream_t stream) {
  const float* x   = (const float*)d_in[0];
  const int*   ei  = (const int*)  d_in[1];
  const float* W1  = (const float*)d_in[2];
  const float* as1 = (const float*)d_in[3];
  const float* ad1 = (const float*)d_in[4];
  const float* b1  = (const float*)d_in[5];
  const float* W2  = (const float*)d_in[6];
  const float* as2 = (const float*)d_in[7];
  const float* ad2 = (const float*)d_in[8];
  const float* b2  = (const float*)d_in[9];

  const int N  = in_sizes[0] / 64;
  const int E  = in_sizes[1] / 2;
  const int EE = E + N;           // with self-loops

  // Workspace layout (floats). Feature buffer and output buffer are each
  // reused across the two layers; total ~ (2*128 + 8)*N + 2*EE floats (~60 MB).
  float* ws    = (float*)d_ws;
  float* h     = ws;                        // N*128 (layer1 uses first N*64)
  float* outA  = h    + (size_t)N * 128;    // N*128 (out1 = first N*64, then out2)
  float* a_src = outA + (size_t)N * 128;    // N*2
  float* a_dst = a_src + (size_t)N * 2;     // N*2
  float* amax  = a_dst + (size_t)N * 2;     // N*2
  float* denom = amax  + (size_t)N * 2;     // N*2
  float* exA   = denom + (size_t)N * 2;     // EE*2

  const int TB = 256;
  auto blk = [](long long work, int tb) { return (int)((work + tb - 1) / tb); };
  const int gemm_grid = blk((long long)((N + 15) / 16) * 32, TB);

  // ======================= layer 1: 64 -> [2,32] concat ====================
  gat_gemm_wmma<64><<<gemm_grid, TB, 0, stream>>>(x, W1, h, N);
  attn_scalars<<<blk(N, TB), TB, 0, stream>>>(h, as1, ad1, a_src, a_dst, N, 32);
  fill_k<<<blk(N * 2, TB), TB, 0, stream>>>(amax, -1e30f, N * 2);
  fill_k<<<blk(N * 2, TB), TB, 0, stream>>>(denom, 0.0f, N * 2);
  edge_max_k<<<blk(EE, TB), TB, 0, stream>>>(ei, a_src, a_dst, amax, E, EE);
  edge_exp_k<<<blk(EE, TB), TB, 0, stream>>>(ei, a_src, a_dst, amax, exA, denom, E, EE);
  init_bias_k<<<blk((long long)N * 64, TB), TB, 0, stream>>>(outA, b1, N, 64);
  edge_aggr_k<64, 32><<<blk((long long)EE * 16, TB), TB, 0, stream>>>(
      ei, h, exA, denom, outA, E, EE);

  // ======================= layer 2: 64 -> [2,64] mean ======================
  gat_gemm_wmma<128><<<gemm_grid, TB, 0, stream>>>(outA, W2, h, N);
  attn_scalars<<<blk(N, TB), TB, 0, stream>>>(h, as2, ad2, a_src, a_dst, N, 64);
  fill_k<<<blk(N * 2, TB), TB, 0, stream>>>(amax, -1e30f, N * 2);
  fill_k<<<blk(N * 2, TB), TB, 0, stream>>>(denom, 0.0f, N * 2);
  edge_max_k<<<blk(EE, TB), TB, 0, stream>>>(ei, a_src, a_dst, amax, E, EE);
  edge_exp_k<<<blk(EE, TB), TB, 0, stream>>>(ei, a_src, a_dst, amax, exA, denom, E, EE);
  fill_k<<<blk((long long)N * 128, TB), TB, 0, stream>>>(outA, 0.0f, N * 128); // out1 dead
  edge_aggr_k<128, 64><<<blk((long long)EE * 32, TB), TB, 0, stream>>>(
      ei, h, exA, denom, outA, E, EE);
  finalize_mean_k<<<blk((long long)N * 64, TB), TB, 0, stream>>>(
      outA, b2, (float*)d_out, N);
}